// GAT_20023137534368
// MI455X (gfx1250) — compile-verified
//
#include <hip/hip_runtime.h>

// GAT (2 layers x 2 graphs) for MI455X / gfx1250, wave32.
// GEMM uses V_WMMA_F32_16X16X4_F32 (exact fp32, matches reference).
// Edge phase: 1 wave per edge, L2-resident gathers + global_atomic_add_f32.

typedef __attribute__((ext_vector_type(2))) float v2f;
typedef __attribute__((ext_vector_type(8))) float v8f;

#define DIM 256

__device__ __forceinline__ void atomAddF(float* p, float v) {
    __hip_atomic_fetch_add(p, v, __ATOMIC_RELAXED, __HIP_MEMORY_SCOPE_AGENT);
}

// ---------------------------------------------------------------- zero fill
__global__ void gat_zero_f32(float* __restrict__ p, long n) {
    long i = (long)blockIdx.x * blockDim.x + threadIdx.x;
    if (i < n) p[i] = 0.0f;
}

// ---------------------------------------------------------------- GEMM (WMMA f32)
// C[N x 256] = A[N x 256] * B[256 x 256], row-major. One 16x16 tile per wave.
// A frag (16x4): lane<16 -> K=0,1 ; lane>=16 -> K=2,3 (ISA 32-bit A layout).
// B frag (4x16): VGPR0 = row K (lane<16) / K+2 (lane>=16); VGPR1 = K+1 / K+3.
// C/D: VGPR r -> (M=r, N=lane) lanes 0-15, (M=r+8, N=lane-16) lanes 16-31.
__global__ void gat_gemm_wmma(const float* __restrict__ A,
                              const float* __restrict__ B,
                              float* __restrict__ C,
                              int N, int nTiles) {
    int lane  = threadIdx.x & 31;
    int tile  = blockIdx.x * (blockDim.x >> 5) + (threadIdx.x >> 5);
    if (tile >= nTiles) return;                 // wave-uniform: EXEC stays all-ones
    int row0  = (tile >> 4) * 16;               // 16 column tiles (256/16)
    int col0  = (tile & 15) * 16;
    if (row0 + 16 > N) return;

    int lhalf = lane >> 4;                      // 0 or 1
    int lcol  = lane & 15;

    v8f acc = {};
    const float* arow = A + (size_t)(row0 + lcol) * DIM;
#pragma unroll 4
    for (int k = 0; k < DIM; k += 4) {
        v2f af = *(const v2f*)(arow + k + 2 * lhalf);
        const float* bp = B + (size_t)(k + 2 * lhalf) * DIM + col0 + lcol;
        v2f bf;
        bf.x = bp[0];
        bf.y = bp[DIM];
        acc = __builtin_amdgcn_wmma_f32_16x16x4_f32(
            false, af, false, bf, (short)0, acc, false, false);
    }

    float* crow = C + (size_t)(row0 + lhalf * 8) * DIM + col0 + lcol;
#pragma unroll
    for (int r = 0; r < 8; ++r) crow[(size_t)r * DIM] = acc[r];
}

// ---------------------------------------------------------------- alpha GEMVs
// One wave per node: alpha_src[i] = hW[i].a[:256], alpha_dst[i] = hW[i].a[256:].
__global__ void gat_alpha(const float* __restrict__ hW,
                          const float* __restrict__ a,
                          float* __restrict__ as, float* __restrict__ ad,
                          int N) {
    int lane = threadIdx.x & 31;
    int i = blockIdx.x * (blockDim.x >> 5) + (threadIdx.x >> 5);
    if (i >= N) return;
    const float* row = hW + (size_t)i * DIM;
    float sl = 0.0f, sd = 0.0f;
#pragma unroll
    for (int j = 0; j < 8; ++j) {
        int c = lane + j * 32;
        float v = row[c];
        sl = fmaf(v, a[c], sl);
        sd = fmaf(v, a[DIM + c], sd);
    }
#pragma unroll
    for (int off = 16; off > 0; off >>= 1) {
        sl += __shfl_xor(sl, off, 32);
        sd += __shfl_xor(sd, off, 32);
    }
    if (lane == 0) { as[i] = sl; ad[i] = sd; }
}

// ---------------------------------------------------------------- edge scatter
// One wave per edge. w = exp(-leakyrelu(as[src]+ad[dst])). Lane l handles
// dims [4l..4l+3] and [128+4l..128+4l+3] via b128 loads + f32 atomics (L2-resident).
__global__ void gat_edges(const float* __restrict__ hW,
                          const float* __restrict__ as,
                          const float* __restrict__ ad,
                          const int* __restrict__ src,
                          const int* __restrict__ dst,
                          float* __restrict__ num,
                          float* __restrict__ rowsum,
                          int E) {
    int lane = threadIdx.x & 31;
    long e = (long)blockIdx.x * (blockDim.x >> 5) + (threadIdx.x >> 5);
    if (e >= E) return;
    int s = src[e];
    int d = dst[e];
    float logit = as[s] + ad[d];
    float lr = logit > 0.0f ? logit : 0.2f * logit;   // leaky_relu, slope 0.2
    float w = __expf(-lr);
    if (lane == 0) atomAddF(&rowsum[s], w);

    const float4* row = (const float4*)(hW + (size_t)d * DIM);
    float4 v0 = row[lane];
    float4 v1 = row[lane + 32];
    float* o = num + (size_t)s * DIM + lane * 4;
    atomAddF(o + 0, w * v0.x);
    atomAddF(o + 1, w * v0.y);
    atomAddF(o + 2, w * v0.z);
    atomAddF(o + 3, w * v0.w);
    float* o2 = o + 128;
    atomAddF(o2 + 0, w * v1.x);
    atomAddF(o2 + 1, w * v1.y);
    atomAddF(o2 + 2, w * v1.z);
    atomAddF(o2 + 3, w * v1.w);
}

// ---------------------------------------------------------------- finalize
// out = elu(num / (rowsum + eps)); D==256 so node index = idx >> 8.
__global__ void gat_finalize(const float* __restrict__ num,
                             const float* __restrict__ rowsum,
                             float* __restrict__ out, long total) {
    long idx = (long)blockIdx.x * blockDim.x + threadIdx.x;
    if (idx >= total) return;
    float r = num[idx] / (rowsum[idx >> 8] + 1e-9f);
    out[idx] = r > 0.0f ? r : expm1f(r);
}

// ---------------------------------------------------------------- host side
static void run_layer(const float* hIn, float* hOut,
                      const float* W, const float* a,
                      const int* src, const int* dst,
                      float* hW, float* num, float* rowsum,
                      float* as, float* ad,
                      int N, int E, hipStream_t stream) {
    long zn = (long)N * DIM + N;                 // num + rowsum (contiguous)
    gat_zero_f32<<<(int)((zn + 255) / 256), 256, 0, stream>>>(num, zn);

    int nTiles = (N / 16) * 16;                  // row tiles * 16 col tiles
    gat_gemm_wmma<<<(nTiles + 3) / 4, 128, 0, stream>>>(hIn, W, hW, N, nTiles);

    gat_alpha<<<(N + 7) / 8, 256, 0, stream>>>(hW, a, as, ad, N);

    gat_edges<<<(E + 7) / 8, 256, 0, stream>>>(hW, as, ad, src, dst,
                                               num, rowsum, E);

    long total = (long)N * DIM;
    gat_finalize<<<(int)((total + 255) / 256), 256, 0, stream>>>(num, rowsum,
                                                                 hOut, total);
}

extern "C" void kernel_launch(void* const* d_in, const int* in_sizes, int n_in,
                              void* d_out, int out_size, void* d_ws, size_t ws_size,
                              hipStream_t stream) {
    const float* sr = (const float*)d_in[0];
    const float* tg = (const float*)d_in[1];
    const float* W1 = (const float*)d_in[2];
    const float* a1 = (const float*)d_in[3];
    const float* W2 = (const float*)d_in[4];
    const float* a2 = (const float*)d_in[5];
    const int* adjS = (const int*)d_in[6];
    const int* adjT = (const int*)d_in[7];

    int N = in_sizes[0] / DIM;
    int E = in_sizes[6] / 2;

    const int* srcS = adjS;       const int* dstS = adjS + E;
    const int* srcT = adjT;       const int* dstT = adjT + E;

    float* out   = (float*)d_out;
    float* outSr = out;
    float* outTg = out + (size_t)N * DIM;

    // Workspace: hW [N*D] | num [N*D] | rowsum [N] | alpha_src [N] | alpha_dst [N]
    float* f      = (float*)d_ws;
    float* hW     = f;
    float* num    = hW + (size_t)N * DIM;
    float* rowsum = num + (size_t)N * DIM;
    float* as     = rowsum + N;
    float* ad     = as + N;

    // Layer 1 writes its (post-ELU) output into d_out halves, which layer 2
    // then consumes; layer 2's finalize overwrites them reading only num/rowsum.
    run_layer(sr,    outSr, W1, a1, srcS, dstS, hW, num, rowsum, as, ad, N, E, stream);
    run_layer(tg,    outTg, W1, a1, srcT, dstT, hW, num, rowsum, as, ad, N, E, stream);
    run_layer(outSr, outSr, W2, a2, srcS, dstS, hW, num, rowsum, as, ad, N, E, stream);
    run_layer(outTg, outTg, W2, a2, srcT, dstT, hW, num, rowsum, as, ad, N, E, stream);
}